// CompGCNConv_86260123173501
// MI455X (gfx1250) — compile-verified
//
#include <hip/hip_runtime.h>
#include <hip/hip_bf16.h>

#define D 256
#define WPB 8          // waves per block (wave32 -> blockDim 256)
#define NT 4           // N-tiles (16 cols each) per wave in the big GEMM
#define BN_ROWS 512
#define TDIM 32

typedef __attribute__((ext_vector_type(2))) float v2f;
typedef __attribute__((ext_vector_type(8))) float v8f;

// ---------------------------------------------------------------- zero ws
__global__ __launch_bounds__(256) void k_zero(float* __restrict__ p, size_t n) {
    size_t i = (size_t)blockIdx.x * blockDim.x + threadIdx.x;
    size_t stride = (size_t)gridDim.x * blockDim.x;
    for (; i < n; i += stride) p[i] = 0.0f;
}

// ------------------------------------------------- LDS-tiled weight transpose
// dst[n*D + k] = src[k*D + n]
__global__ __launch_bounds__(256) void k_transpose(const float* __restrict__ src,
                                                   float* __restrict__ dst) {
    __shared__ float tile[TDIM][TDIM + 1];      // +1 pad: no bank conflicts
    const int bx = blockIdx.x % (D / TDIM);     // column-block of src
    const int by = blockIdx.x / (D / TDIM);     // row-block of src
    const int tx = threadIdx.x & 31;
    const int ty = threadIdx.x >> 5;            // 0..7
#pragma unroll
    for (int j = 0; j < TDIM; j += 8)
        tile[ty + j][tx] = src[(size_t)(by * TDIM + ty + j) * D + bx * TDIM + tx];
    __syncthreads();
#pragma unroll
    for (int j = 0; j < TDIM; j += 8)
        dst[(size_t)(bx * TDIM + ty + j) * D + by * TDIM + tx] = tile[tx][ty + j];
}

// ---------------------------------------------------------------- degrees
__global__ __launch_bounds__(256) void k_degrees(const int* __restrict__ eidx,
                                                 float* __restrict__ deg_in,
                                                 float* __restrict__ deg_out,
                                                 int E2, int E) {
    int e = blockIdx.x * blockDim.x + threadIdx.x;
    if (e >= E2) return;
    int dst = eidx[e];                          // edge_index[0][e] == row
    unsafeAtomicAdd((e < E) ? &deg_in[dst] : &deg_out[dst], 1.0f);
}

__global__ __launch_bounds__(256) void k_deginv(float* __restrict__ deg_in,
                                                float* __restrict__ deg_out,
                                                int n_ent) {
    int i = blockIdx.x * blockDim.x + threadIdx.x;
    if (i >= n_ent) return;
    float a = deg_in[i];
    float b = deg_out[i];
    deg_in[i]  = (a > 0.0f) ? rsqrtf(a) : 0.0f;
    deg_out[i] = (b > 0.0f) ? rsqrtf(b) : 0.0f;
}

// ------------------------------------------------- edge scatter (one wave/edge)
__global__ __launch_bounds__(256) void k_scatter(const float* __restrict__ x,
                                                 const int* __restrict__ eidx,
                                                 const int* __restrict__ etype,
                                                 const float* __restrict__ rel,
                                                 const float* __restrict__ dinv_in,
                                                 const float* __restrict__ dinv_out,
                                                 float* __restrict__ agg_in,
                                                 float* __restrict__ agg_out,
                                                 int E2, int E) {
    int e = blockIdx.x * WPB + (threadIdx.x >> 5);
    if (e >= E2) return;
    int lane = threadIdx.x & 31;

    int dst = eidx[e];            // row (destination / segment id)
    int src = eidx[E2 + e];       // col (source node j)
    int t   = etype[e];
    bool isin = (e < E);

    const float* dinv = isin ? dinv_in : dinv_out;
    float norm = dinv[dst] * dinv[src];

    const float* xs = x   + (size_t)src * D;
    const float* rl = rel + (size_t)t   * D;
    float* ag = (isin ? agg_in : agg_out) + (size_t)dst * D;

#pragma unroll
    for (int j = 0; j < D / 32; ++j) {
        int c = lane + 32 * j;
        unsafeAtomicAdd(&ag[c], xs[c] * rl[c] * norm);
    }
}

// ---------------- fused GEMM: one wave -> 16x64 out strip (4 accumulators)
// out = (agg_in@W_in + agg_out@W_out + (x*loop_rel)@W_loop) / 3 + bias
// Weights are pre-transposed: wT[n*D + k] = W[k*D + n]  ->  B frag = one b64 load
__global__ __launch_bounds__(256) void k_gemm(const float* __restrict__ agg_in,
                                              const float* __restrict__ agg_out,
                                              const float* __restrict__ x,
                                              const float* __restrict__ loop_rel,
                                              const float* __restrict__ w_inT,
                                              const float* __restrict__ w_outT,
                                              const float* __restrict__ w_loopT,
                                              const float* __restrict__ bias,
                                              float* __restrict__ out,
                                              int n_ent) {
    const int wave  = blockIdx.x * WPB + (threadIdx.x >> 5);
    const int mtile = wave >> 2;              // 4 waves per M-strip (N=256 = 4*64)
    const int ngrp  = wave & 3;
    if (mtile * 16 >= n_ent) return;          // wave-uniform guard (EXEC stays full)

    const int lane  = threadIdx.x & 31;
    const int l16   = lane & 15;
    const int khalf = lane >> 4;              // 0 or 1
    const int m     = mtile * 16 + l16;       // A row for this lane
    const int n0    = ngrp * 64 + l16;        // first B/D column for this lane

    const float* pa_in  = agg_in  + (size_t)m * D;
    const float* pa_out = agg_out + (size_t)m * D;
    const float* pa_x   = x       + (size_t)m * D;

    v8f acc[NT];
#pragma unroll
    for (int t = 0; t < NT; ++t) acc[t] = (v8f){0.f,0.f,0.f,0.f,0.f,0.f,0.f,0.f};

#pragma unroll 2
    for (int kk = 0; kk < D; kk += 4) {
        const int ka = kk + 2 * khalf;        // this lane's K pair (8B aligned)

        v2f ai = *(const v2f*)(pa_in + ka);
        v2f ao = *(const v2f*)(pa_out + ka);
        v2f lr = *(const v2f*)(loop_rel + ka);
        v2f ax = *(const v2f*)(pa_x + ka);
        v2f al = { ax.x * lr.x, ax.y * lr.y };

        // matrix-major, tile-minor: 4 insts between same-accumulator reuses
#pragma unroll
        for (int t = 0; t < NT; ++t) {
            v2f b = *(const v2f*)(w_inT + (size_t)(n0 + 16 * t) * D + ka);
            acc[t] = __builtin_amdgcn_wmma_f32_16x16x4_f32(false, ai, false, b,
                                                           (short)0, acc[t], false, false);
        }
#pragma unroll
        for (int t = 0; t < NT; ++t) {
            v2f b = *(const v2f*)(w_outT + (size_t)(n0 + 16 * t) * D + ka);
            acc[t] = __builtin_amdgcn_wmma_f32_16x16x4_f32(false, ao, false, b,
                                                           (short)0, acc[t], false, false);
        }
#pragma unroll
        for (int t = 0; t < NT; ++t) {
            v2f b = *(const v2f*)(w_loopT + (size_t)(n0 + 16 * t) * D + ka);
            acc[t] = __builtin_amdgcn_wmma_f32_16x16x4_f32(false, al, false, b,
                                                           (short)0, acc[t], false, false);
        }
    }

#pragma unroll
    for (int t = 0; t < NT; ++t) {
        const int n = n0 + 16 * t;
        const float bcol = bias[n];
#pragma unroll
        for (int v = 0; v < 8; ++v) {
            int row = mtile * 16 + v + 8 * khalf;  // C/D layout: VGPR v -> M=v / v+8
            out[(size_t)row * D + n] = acc[t][v] * (1.0f / 3.0f) + bcol;
        }
    }
}

// ---------------------------------------------------------------- BN stats
__global__ __launch_bounds__(256) void k_bn_reduce(const float* __restrict__ out,
                                                   float* __restrict__ bn_sum,
                                                   float* __restrict__ bn_sumsq,
                                                   int n_ent) {
    int c  = threadIdx.x;                     // one channel per thread
    int r0 = blockIdx.x * BN_ROWS;
    int r1 = min(r0 + BN_ROWS, n_ent);
    float s = 0.f, s2 = 0.f;
    for (int r = r0; r < r1; ++r) {
        float v = out[(size_t)r * D + c];     // coalesced across threads
        s += v; s2 += v * v;
    }
    unsafeAtomicAdd(&bn_sum[c], s);
    unsafeAtomicAdd(&bn_sumsq[c], s2);
}

__global__ __launch_bounds__(256) void k_bn_final(const float* __restrict__ bn_sum,
                                                  const float* __restrict__ bn_sumsq,
                                                  const float* __restrict__ gamma,
                                                  const float* __restrict__ beta,
                                                  float* __restrict__ scale,
                                                  float* __restrict__ shift,
                                                  int n_ent) {
    int c = threadIdx.x;
    float inv_n = 1.0f / (float)n_ent;
    float mean  = bn_sum[c] * inv_n;
    float var   = bn_sumsq[c] * inv_n - mean * mean;
    float s     = gamma[c] * rsqrtf(var + 1e-5f);
    scale[c] = s;
    shift[c] = beta[c] - mean * s;
}

__global__ __launch_bounds__(256) void k_bn_apply(float* __restrict__ out,
                                                  const float* __restrict__ scale,
                                                  const float* __restrict__ shift,
                                                  size_t n4) {
    size_t i = (size_t)blockIdx.x * blockDim.x + threadIdx.x;
    if (i >= n4) return;
    float4 v = ((float4*)out)[i];
    int c = (int)((i * 4) & (D - 1));
    v.x = v.x * scale[c + 0] + shift[c + 0];
    v.y = v.y * scale[c + 1] + shift[c + 1];
    v.z = v.z * scale[c + 2] + shift[c + 2];
    v.w = v.w * scale[c + 3] + shift[c + 3];
    ((float4*)out)[i] = v;
}

// ------------------------------------------- rel_out = rel_embed @ w_rel (200x256)
__global__ __launch_bounds__(256) void k_relgemm(const float* __restrict__ rel,
                                                 const float* __restrict__ w_relT,
                                                 float* __restrict__ rel_out,
                                                 int n_rel) {
    const int wave  = blockIdx.x * WPB + (threadIdx.x >> 5);
    const int mtile = wave >> 4;
    const int ntile = wave & 15;
    if (mtile * 16 >= n_rel + 15) return;     // wave-uniform

    const int lane  = threadIdx.x & 31;
    const int l16   = lane & 15;
    const int khalf = lane >> 4;
    const int m     = mtile * 16 + l16;
    const int n     = ntile * 16 + l16;

    // branch-free OOB handling so EXEC stays all-ones through WMMA
    const int   mc  = (m < n_rel) ? m : 0;
    const float msk = (m < n_rel) ? 1.0f : 0.0f;
    const float* pa = rel + (size_t)mc * D;
    const float* pb = w_relT + (size_t)n * D;

    v8f c = {0.f, 0.f, 0.f, 0.f, 0.f, 0.f, 0.f, 0.f};

#pragma unroll 4
    for (int kk = 0; kk < D; kk += 4) {
        const int ka = kk + 2 * khalf;
        v2f a = *(const v2f*)(pa + ka);
        a.x *= msk; a.y *= msk;
        v2f b = *(const v2f*)(pb + ka);
        c = __builtin_amdgcn_wmma_f32_16x16x4_f32(false, a, false, b,
                                                  (short)0, c, false, false);
    }

#pragma unroll
    for (int v = 0; v < 8; ++v) {
        int row = mtile * 16 + v + 8 * khalf;
        if (row < n_rel) rel_out[(size_t)row * D + n] = c[v];
    }
}

// ---------------------------------------------------------------- launcher
extern "C" void kernel_launch(void* const* d_in, const int* in_sizes, int n_in,
                              void* d_out, int out_size, void* d_ws, size_t ws_size,
                              hipStream_t stream) {
    const float* x        = (const float*)d_in[0];
    const int*   eidx     = (const int*)  d_in[1];
    const int*   etype    = (const int*)  d_in[2];
    const float* rel      = (const float*)d_in[3];
    const float* w_loop   = (const float*)d_in[4];
    const float* w_in     = (const float*)d_in[5];
    const float* w_out    = (const float*)d_in[6];
    const float* w_rel    = (const float*)d_in[7];
    const float* loop_rel = (const float*)d_in[8];
    const float* bias     = (const float*)d_in[9];
    const float* gamma    = (const float*)d_in[10];
    const float* beta     = (const float*)d_in[11];

    const int n_ent = in_sizes[0] / D;        // 100000
    const int E2    = in_sizes[2];            // 1,000,000
    const int E     = E2 / 2;                 // 500,000
    const int n_rel = in_sizes[3] / D;        // 200

    float* out     = (float*)d_out;
    float* rel_out = out + (size_t)n_ent * D;

    // workspace layout (zeroed region first)
    float* ws       = (float*)d_ws;
    float* agg_in   = ws;
    float* agg_out  = agg_in  + (size_t)n_ent * D;
    float* deg_in   = agg_out + (size_t)n_ent * D;
    float* deg_out  = deg_in  + n_ent;
    float* bn_sum   = deg_out + n_ent;
    float* bn_sumsq = bn_sum  + D;
    float* bn_scale = bn_sumsq + D;
    float* bn_shift = bn_scale + D;
    float* w_inT    = bn_shift + D;           // fully overwritten, no zeroing
    float* w_outT   = w_inT   + (size_t)D * D;
    float* w_loopT  = w_outT  + (size_t)D * D;
    float* w_relT   = w_loopT + (size_t)D * D;

    // 1) zero accumulators (agg_in, agg_out, deg_in, deg_out, bn_sum, bn_sumsq)
    size_t zero_n = (size_t)2 * n_ent * D + 2 * (size_t)n_ent + 2 * D;
    k_zero<<<2048, 256, 0, stream>>>(ws, zero_n);

    // 2) transpose the four weight matrices (LDS-tiled)
    const int tp_blocks = (D / TDIM) * (D / TDIM);  // 64
    k_transpose<<<tp_blocks, 256, 0, stream>>>(w_in,   w_inT);
    k_transpose<<<tp_blocks, 256, 0, stream>>>(w_out,  w_outT);
    k_transpose<<<tp_blocks, 256, 0, stream>>>(w_loop, w_loopT);
    k_transpose<<<tp_blocks, 256, 0, stream>>>(w_rel,  w_relT);

    // 3) degrees + deg^-0.5
    k_degrees<<<(E2 + 255) / 256, 256, 0, stream>>>(eidx, deg_in, deg_out, E2, E);
    k_deginv<<<(n_ent + 255) / 256, 256, 0, stream>>>(deg_in, deg_out, n_ent);

    // 4) edge scatter-aggregate (one wave per edge)
    k_scatter<<<(E2 + WPB - 1) / WPB, 32 * WPB, 0, stream>>>(
        x, eidx, etype, rel, deg_in, deg_out, agg_in, agg_out, E2, E);

    // 5) fused WMMA GEMM -> pre-BN output (one wave per 16x64 strip)
    int mtiles = (n_ent + 15) / 16;
    int waves  = mtiles * NT;
    k_gemm<<<(waves + WPB - 1) / WPB, 32 * WPB, 0, stream>>>(
        agg_in, agg_out, x, loop_rel, w_inT, w_outT, w_loopT, bias, out, n_ent);

    // 6) batch-norm
    k_bn_reduce<<<(n_ent + BN_ROWS - 1) / BN_ROWS, 256, 0, stream>>>(
        out, bn_sum, bn_sumsq, n_ent);
    k_bn_final<<<1, 256, 0, stream>>>(bn_sum, bn_sumsq, gamma, beta,
                                      bn_scale, bn_shift, n_ent);
    size_t n4 = (size_t)n_ent * D / 4;
    k_bn_apply<<<(int)((n4 + 255) / 256), 256, 0, stream>>>(out, bn_scale, bn_shift, n4);

    // 7) rel_out = rel_embed @ w_rel
    int rmt    = (n_rel + 15) / 16;
    int rwaves = rmt * 16;
    k_relgemm<<<(rwaves + WPB - 1) / WPB, 32 * WPB, 0, stream>>>(rel, w_relT, rel_out, n_rel);
}